// QuanvolutionClassifier_65481071407224
// MI455X (gfx1250) — compile-verified
//
#include <hip/hip_runtime.h>
#include <math.h>

typedef __attribute__((ext_vector_type(2))) float v2f;
typedef __attribute__((ext_vector_type(8))) float v8f;

#define WAVES_PER_BLOCK 4
#define PATCHES 196
#define PATCH_PER_WAVE (PATCHES / WAVES_PER_BLOCK) /* 49 */

// Fused quanvolution features + [B,784]x[784,16] fp32 WMMA GEMM + log_softmax.
// theta provably cancels (RZ phases have unit modulus; CNOTs permute basis
// states; probabilities are |amp|^2), so features reduce to cos products.
__global__ __launch_bounds__(WAVES_PER_BLOCK * 32)
void quanv_wmma_fused(const float* __restrict__ x,
                      const float* __restrict__ Wm,
                      const float* __restrict__ bias,
                      float* __restrict__ out, int B)
{
    __shared__ float red[WAVES_PER_BLOCK][32][8];

    const int lane = threadIdx.x & 31;
    const int wave = threadIdx.x >> 5;
    const int rowBase = blockIdx.x * 16;
    const int m  = lane & 15;   // A: row M within tile; B: column N (class id)
    const int hi = lane >> 4;   // 0: K-offsets {0,1} / patch top row; 1: {2,3} / bottom row

    int arow = rowBase + m;
    if (arow >= B) arow = B - 1;                 // clamp (stores guarded below)
    const float* xrow = x + (size_t)arow * 784;
    const bool nvalid = (m < 10);
    const float* wrow = Wm + (size_t)m * 784;    // W[n, :], n = m

    v8f acc = {0.f, 0.f, 0.f, 0.f, 0.f, 0.f, 0.f, 0.f};

    const int p0 = wave * PATCH_PER_WAVE;
    #pragma unroll 7
    for (int i = 0; i < PATCH_PER_WAVE; ++i) {
        const int p  = p0 + i;                   // patch index = K-tile index
        const int pr = p / 14;
        const int pc = p - pr * 14;
        // two adjacent pixels of the 2x2 patch (top row for hi=0, bottom for hi=1)
        const int pix = (2 * pr + hi) * 28 + 2 * pc;   // even -> 8B aligned
        const v2f xv = *reinterpret_cast<const v2f*>(xrow + pix);
        const float ca = __cosf(xv.x);
        v2f a;
        a.x = ca;                 // feature 4p + 2*hi     : cos(x_first)
        a.y = ca * __cosf(xv.y);  // feature 4p + 2*hi + 1 : cos(x_first)*cos(x_second)
        v2f bv = {0.f, 0.f};
        if (nvalid)               // B fragment: W^T[k, n], zero-padded for n>=10
            bv = *reinterpret_cast<const v2f*>(wrow + 4 * p + 2 * hi);
        acc = __builtin_amdgcn_wmma_f32_16x16x4_f32(
                  /*neg_a=*/false, a, /*neg_b=*/false, bv,
                  /*c_mod=*/(short)0, acc, /*reuse_a=*/false, /*reuse_b=*/false);
    }

    // K-split reduction across the 4 waves of this block.
    #pragma unroll
    for (int r = 0; r < 8; ++r) red[wave][lane][r] = acc[r];
    __syncthreads();
    if (wave != 0) return;

    #pragma unroll
    for (int r = 0; r < 8; ++r)
        acc[r] = red[0][lane][r] + red[1][lane][r] + red[2][lane][r] + red[3][lane][r];

    const float bn = nvalid ? bias[m] : 0.0f;

    // C/D layout: VGPR r -> (M = r, N = lane) for lanes 0-15, (M = r+8) for 16-31.
    // log_softmax over the 10 valid classes via 16-lane shuffle reductions.
    #pragma unroll
    for (int r = 0; r < 8; ++r) {
        const int row = rowBase + r + hi * 8;
        const float v = acc[r] + bn;
        float vm = nvalid ? v : -3.402823466e38f;
        #pragma unroll
        for (int off = 8; off >= 1; off >>= 1)
            vm = fmaxf(vm, __shfl_xor(vm, off, 16));
        float e = nvalid ? __expf(v - vm) : 0.0f;
        #pragma unroll
        for (int off = 8; off >= 1; off >>= 1)
            e += __shfl_xor(e, off, 16);
        const float res = v - vm - __logf(e);
        if (nvalid && row < B)
            out[(size_t)row * 10 + m] = res;
    }
}

extern "C" void kernel_launch(void* const* d_in, const int* in_sizes, int n_in,
                              void* d_out, int out_size, void* d_ws, size_t ws_size,
                              hipStream_t stream) {
    (void)n_in; (void)out_size; (void)d_ws; (void)ws_size;
    const float* x    = (const float*)d_in[0];
    // d_in[1] = theta: provably has no effect on the output (phases cancel in |.|^2)
    const float* Wm   = (const float*)d_in[2];
    const float* bias = (const float*)d_in[3];
    float* out = (float*)d_out;

    const int B = in_sizes[0] / 784;
    const int tiles = (B + 15) / 16;   // one 16-row output tile per block
    dim3 grid(tiles), block(WAVES_PER_BLOCK * 32);
    hipLaunchKernelGGL(quanv_wmma_fused, grid, block, 0, stream, x, Wm, bias, out, B);
}